// NequipBlock_79508434583959
// MI455X (gfx1250) — compile-verified
//
#include <hip/hip_runtime.h>

typedef float v2f __attribute__((ext_vector_type(2)));
typedef float v8f __attribute__((ext_vector_type(8)));

#define N_NODES 50000
#define N_EDGES 800000
#define MULQ    64
#define NSPQ    5

#if defined(__HIP_DEVICE_COMPILE__)
#if !__has_builtin(__builtin_amdgcn_wmma_f32_16x16x4_f32)
#error "DEVICE pass: __builtin_amdgcn_wmma_f32_16x16x4_f32 not available"
#endif
#endif

__device__ __forceinline__ float silu_f(float x) {
#if defined(__HIP_DEVICE_COMPILE__)
    // x * 1/(1+e^-x); v_rcp_f32 is ~1ulp, avoids the IEEE div sequence
    return x * __builtin_amdgcn_rcpf(1.0f + __expf(-x));
#else
    return x / (1.0f + expf(-x)); // host pass stub; never executed
#endif
}

__device__ __forceinline__ v8f wmma4(v2f a, v2f b, v8f c) {
#if defined(__HIP_DEVICE_COMPILE__)
    // D = A(16x4 f32) * B(4x16 f32) + C(16x16 f32)
    return __builtin_amdgcn_wmma_f32_16x16x4_f32(
        false, a, false, b, (short)0, c, false, false);
#else
    (void)a; (void)b;
    return c; // host pass stub; never executed
#endif
}

__device__ __forceinline__ void wavebar() {
#if defined(__HIP_DEVICE_COMPILE__)
    __builtin_amdgcn_wave_barrier();
#endif
}

// Load a 16x4 fragment (A) or 4x16 fragment (B, from a transposed [n][k] table).
// Row = lane&15 ; K-pair = k0 + 2*(lane>>4). All offsets even -> b64 LDS loads.
__device__ __forceinline__ v2f load_frag(const float* buf, int stride, int k0, int lane) {
    int row = lane & 15;
    int dk  = (lane >> 4) << 1;
    const float2* p = (const float2*)(buf + row * stride + k0 + dk);
    float2 t = *p;
    v2f a; a.x = t.x; a.y = t.y;
    return a;
}

// ---------------------------------------------------------------- zero scratch
__global__ void zero_kernel(float* __restrict__ p, int n) {
    int i = blockIdx.x * blockDim.x + threadIdx.x;
    if (i < n) p[i] = 0.0f;
}

// ------------------------------------------- per-species tables (res1, x0l)
// x0[sp]   = W_embed[sp]/sqrt(5)
// res1[sp] = (1/sqrt(320)) * x0[sp] @ W_res1[:,sp,:]
// x0l[sp]  = (1/8)        * x0[sp] @ W_lin_in
__global__ void precompute_kernel(const float* __restrict__ W_embed,
                                  const float* __restrict__ W_res1,
                                  const float* __restrict__ W_lin_in,
                                  float* __restrict__ res1_sp,
                                  float* __restrict__ x0l_sp) {
    int i = blockIdx.x * blockDim.x + threadIdx.x;
    if (i >= NSPQ * MULQ) return;
    int sp = i >> 6;
    int w  = i & 63;
    float r = 0.0f, x = 0.0f;
    for (int u = 0; u < MULQ; ++u) {
        float e = W_embed[sp * MULQ + u] * 0.44721359549995794f; // /sqrt(5)
        r = fmaf(e, W_res1[(u * NSPQ + sp) * MULQ + w], r);
        x = fmaf(e, W_lin_in[u * MULQ + w], x);
    }
    res1_sp[i] = r * 0.05590169943749474f;  // 1/sqrt(320)
    x0l_sp[i]  = x * 0.125f;                // 1/sqrt(64)
}

// --------------------------------------------------------------- edge kernel
// Per 16-edge tile (one wave): radial[16x8] -> MLP 8->64->64->64 via
// V_WMMA_F32_16X16X4_F32, then e0 = w0 * x0l[spec(sender)] / 16 scattered
// into n0[receiver] with global_atomic_add_f32.
// All per-layer scale constants are folded into the LDS weight tables.
__global__ __launch_bounds__(128) void edge_kernel(
        const float* __restrict__ edge_vectors,
        const int*   __restrict__ node_species,
        const int*   __restrict__ senders,
        const int*   __restrict__ receivers,
        const float* __restrict__ W_r1,
        const float* __restrict__ W_r2,
        const float* __restrict__ W_r3,
        const float* __restrict__ x0l_sp,
        float*       __restrict__ n0) {
    __shared__ __align__(16) float Wr1t[64 * 8];    // [n][k] * 1/sqrt(8)
    __shared__ __align__(16) float Wr2t[64 * 66];   // [n][k] * 1/8
    __shared__ __align__(16) float Wr3t[64 * 66];   // cols 0..63, [n][k] * 1/128
    __shared__ __align__(16) float x0l_lds[NSPQ * MULQ];
    __shared__ __align__(16) float hbuf_all[4][16 * 66]; // per-wave activations
    __shared__ int sps_all[4][16];
    __shared__ int rcv_all[4][16];

    const int tid = threadIdx.x;
    for (int i = tid; i < 64 * 64; i += 128) {
        int k = i >> 6, n = i & 63;
        Wr2t[n * 66 + k] = W_r2[i] * 0.125f;               // fold 1/sqrt(H)
        // fold 1/8 (w-scale) * 1/16 (AVG_NEI) into layer-3 weights
        Wr3t[n * 66 + k] = W_r3[k * 128 + n] * 0.0078125f;
    }
    for (int i = tid; i < 8 * 64; i += 128) {
        int k = i >> 6, n = i & 63;
        Wr1t[n * 8 + k] = W_r1[i] * 0.3535533905932738f;   // fold 1/sqrt(NB)
    }
    for (int i = tid; i < NSPQ * MULQ; i += 128) x0l_lds[i] = x0l_sp[i];
    __syncthreads();

    const int wave = tid >> 5;
    const int lane = tid & 31;
    float* hbuf = hbuf_all[wave];
    int* sps = sps_all[wave];
    int* rcv = rcv_all[wave];

    const int gwave  = blockIdx.x * 4 + wave;
    const int nwaves = gridDim.x * 4;
    const int ntiles = N_EDGES / 16;

    for (int t = gwave; t < ntiles; t += nwaves) {
        const int ebase = t * 16;

        // ---- meta + radial features for 16 edges (lanes 0..15)
        if (lane < 16) {
            int e = ebase + lane;
            int s = senders[e];
            sps[lane] = node_species[s];
            rcv[lane] = receivers[e];
            float vx = edge_vectors[3 * e + 0];
            float vy = edge_vectors[3 * e + 1];
            float vz = edge_vectors[3 * e + 2];
            float r2 = vx * vx + vy * vy + vz * vz;
            float r  = sqrtf(fmaxf(r2, 1e-12f));
            float x  = r * 0.2f;                 // r / RMAX
            float env = 0.0f;
            if (x < 1.0f) {
                float x2 = x * x;
                float x5 = x2 * x2 * x;
                env = 1.0f - 21.0f * x5 + 35.0f * x5 * x - 15.0f * x5 * x2;
            }
            float pref = 0.6324555320336759f / r * env; // sqrt(2/5)/r * env
            float arg  = 3.14159265358979323846f * x;   // pi*r/RMAX
            #pragma unroll
            for (int nb = 0; nb < 8; ++nb)
                hbuf[lane * 66 + nb] = pref * sinf((float)(nb + 1) * arg);
        }
        wavebar();

        // ---- layer 1: rad[16x8] @ (W_r1/sqrt(8))[8x64], silu
        v8f acc1[4] = {};
        #pragma unroll
        for (int k0 = 0; k0 < 8; k0 += 4) {
            v2f a = load_frag(hbuf, 66, k0, lane);
            #pragma unroll
            for (int nt = 0; nt < 4; ++nt) {
                v2f b = load_frag(Wr1t + nt * 16 * 8, 8, k0, lane);
                acc1[nt] = wmma4(a, b, acc1[nt]);
            }
        }
        wavebar();
        {
            int n = lane & 15, moff = (lane >> 4) << 3;
            #pragma unroll
            for (int nt = 0; nt < 4; ++nt)
                #pragma unroll
                for (int r = 0; r < 8; ++r)
                    hbuf[(r + moff) * 66 + nt * 16 + n] = silu_f(acc1[nt][r]);
        }
        wavebar();

        // ---- layer 2: h[16x64] @ (W_r2/8)[64x64], silu
        v8f acc2[4] = {};
        #pragma unroll
        for (int k0 = 0; k0 < 64; k0 += 4) {
            v2f a = load_frag(hbuf, 66, k0, lane);
            #pragma unroll
            for (int nt = 0; nt < 4; ++nt) {
                v2f b = load_frag(Wr2t + nt * 16 * 66, 66, k0, lane);
                acc2[nt] = wmma4(a, b, acc2[nt]);
            }
        }
        wavebar();
        {
            int n = lane & 15, moff = (lane >> 4) << 3;
            #pragma unroll
            for (int nt = 0; nt < 4; ++nt)
                #pragma unroll
                for (int r = 0; r < 8; ++r)
                    hbuf[(r + moff) * 66 + nt * 16 + n] = silu_f(acc2[nt][r]);
        }
        wavebar();

        // ---- layer 3: h[16x64] @ (W_r3[:, :64]/128)  -> w0/16
        v8f acc3[4] = {};
        #pragma unroll
        for (int k0 = 0; k0 < 64; k0 += 4) {
            v2f a = load_frag(hbuf, 66, k0, lane);
            #pragma unroll
            for (int nt = 0; nt < 4; ++nt) {
                v2f b = load_frag(Wr3t + nt * 16 * 66, 66, k0, lane);
                acc3[nt] = wmma4(a, b, acc3[nt]);
            }
        }

        // ---- e0 = w0 * x0l[spec(sender)] (scales pre-folded), scatter-add
        {
            int n = lane & 15, moff = (lane >> 4) << 3;
            #pragma unroll
            for (int r = 0; r < 8; ++r) {
                int m   = r + moff;
                int sp  = sps[m];
                int rec = rcv[m];
                float* dst        = n0 + (size_t)rec * MULQ + n;
                const float* xl   = x0l_lds + sp * MULQ + n;
                #pragma unroll
                for (int nt = 0; nt < 4; ++nt) {
                    float val = acc3[nt][r] * xl[nt * 16];
                    atomicAdd(dst + nt * 16, val);
                }
            }
        }
        wavebar();
    }
}

// --------------------------------------------------------------- node kernel
// out_s = silu(n0 @ W_out_s1/8 + res1_sp[spec]); s2f = out_s @ W_o1/8;
// energy = s2f . W_o2 / sqrt(32). Scales folded into LDS weight tables.
__global__ __launch_bounds__(128) void node_kernel(
        const int*   __restrict__ node_species,
        const float* __restrict__ W_out_s1,
        const float* __restrict__ W_o1,
        const float* __restrict__ W_o2,
        const float* __restrict__ res1_sp,
        const float* __restrict__ n0,
        float*       __restrict__ out) {
    __shared__ __align__(16) float Ws1t[64 * 66];   // [n][k] * 1/8
    __shared__ __align__(16) float Wo1t[32 * 66];   // [n][k] * 1/8
    __shared__ __align__(16) float res1_lds[NSPQ * MULQ];
    __shared__ __align__(16) float wo2_lds[32];
    __shared__ __align__(16) float nbuf_all[4][16 * 66];
    __shared__ __align__(16) float sbuf_all[4][16 * 34];
    __shared__ int spb_all[4][16];

    const int tid = threadIdx.x;
    for (int i = tid; i < 64 * 64; i += 128) {
        int k = i >> 6, n = i & 63;
        Ws1t[n * 66 + k] = W_out_s1[i] * 0.125f;   // fold 1/sqrt(MUL)
    }
    for (int i = tid; i < 64 * 32; i += 128) {
        int k = i >> 5, n = i & 31;
        Wo1t[n * 66 + k] = W_o1[i] * 0.125f;       // fold 1/sqrt(MUL)
    }
    for (int i = tid; i < NSPQ * MULQ; i += 128) res1_lds[i] = res1_sp[i];
    if (tid < 32) wo2_lds[tid] = W_o2[tid];
    __syncthreads();

    const int wave = tid >> 5;
    const int lane = tid & 31;
    float* nbuf = nbuf_all[wave];
    float* sbuf = sbuf_all[wave];
    int* spb = spb_all[wave];

    const int gwave  = blockIdx.x * 4 + wave;
    const int nwaves = gridDim.x * 4;
    const int ntiles = N_NODES / 16;

    for (int t = gwave; t < ntiles; t += nwaves) {
        const int nbase = t * 16;

        #pragma unroll
        for (int i = 0; i < 32; ++i) {
            int idx = lane + i * 32;
            int m = idx >> 6, c = idx & 63;
            nbuf[m * 66 + c] = n0[(size_t)(nbase + m) * MULQ + c];
        }
        if (lane < 16) spb[lane] = node_species[nbase + lane];
        wavebar();

        // n0 @ (W_out_s1/8)
        v8f acc[4] = {};
        #pragma unroll
        for (int k0 = 0; k0 < 64; k0 += 4) {
            v2f a = load_frag(nbuf, 66, k0, lane);
            #pragma unroll
            for (int nt = 0; nt < 4; ++nt) {
                v2f b = load_frag(Ws1t + nt * 16 * 66, 66, k0, lane);
                acc[nt] = wmma4(a, b, acc[nt]);
            }
        }
        wavebar();
        {
            int n = lane & 15, moff = (lane >> 4) << 3;
            #pragma unroll
            for (int r = 0; r < 8; ++r) {
                int m = r + moff;
                int sp = spb[m];
                #pragma unroll
                for (int nt = 0; nt < 4; ++nt) {
                    float v = acc[nt][r] + res1_lds[sp * MULQ + nt * 16 + n];
                    nbuf[m * 66 + nt * 16 + n] = silu_f(v);
                }
            }
        }
        wavebar();

        // out_s @ (W_o1/8) (N = 32)
        v8f accB[2] = {};
        #pragma unroll
        for (int k0 = 0; k0 < 64; k0 += 4) {
            v2f a = load_frag(nbuf, 66, k0, lane);
            #pragma unroll
            for (int nt = 0; nt < 2; ++nt) {
                v2f b = load_frag(Wo1t + nt * 16 * 66, 66, k0, lane);
                accB[nt] = wmma4(a, b, accB[nt]);
            }
        }
        wavebar();
        {
            int n = lane & 15, moff = (lane >> 4) << 3;
            #pragma unroll
            for (int nt = 0; nt < 2; ++nt)
                #pragma unroll
                for (int r = 0; r < 8; ++r)
                    sbuf[(r + moff) * 34 + nt * 16 + n] = accB[nt][r];
        }
        wavebar();

        if (lane < 16) {
            float e = 0.0f;
            #pragma unroll
            for (int j = 0; j < 32; ++j)
                e = fmaf(sbuf[lane * 34 + j], wo2_lds[j], e);
            out[nbase + lane] = e * 0.17677669529663687f; // 1/sqrt(32)
        }
        wavebar();
    }
}

extern "C" void kernel_launch(void* const* d_in, const int* in_sizes, int n_in,
                              void* d_out, int out_size, void* d_ws, size_t ws_size,
                              hipStream_t stream) {
    const float* edge_vectors = (const float*)d_in[0];
    const int*   node_species = (const int*)d_in[1];
    const int*   senders      = (const int*)d_in[2];
    const int*   receivers    = (const int*)d_in[3];
    const float* W_embed      = (const float*)d_in[4];
    const float* W_res1       = (const float*)d_in[5];
    // d_in[6] = W_res2   : dead (out_v deleted in reference)
    const float* W_lin_in     = (const float*)d_in[7];
    const float* W_r1         = (const float*)d_in[8];
    const float* W_r2         = (const float*)d_in[9];
    const float* W_r3         = (const float*)d_in[10];
    const float* W_out_s1     = (const float*)d_in[11];
    // d_in[12] = W_out_s2: dead, d_in[13] = W_out_v: dead
    const float* W_o1         = (const float*)d_in[14];
    const float* W_o2         = (const float*)d_in[15];
    float* out = (float*)d_out;

    float* res1_sp = (float*)d_ws;
    float* x0l_sp  = res1_sp + NSPQ * MULQ;
    float* n0      = x0l_sp  + NSPQ * MULQ;

    const int n0_elems = N_NODES * MULQ;
    zero_kernel<<<(n0_elems + 255) / 256, 256, 0, stream>>>(n0, n0_elems);
    precompute_kernel<<<2, 256, 0, stream>>>(W_embed, W_res1, W_lin_in,
                                             res1_sp, x0l_sp);
    edge_kernel<<<512, 128, 0, stream>>>(edge_vectors, node_species, senders,
                                         receivers, W_r1, W_r2, W_r3,
                                         x0l_sp, n0);
    node_kernel<<<256, 128, 0, stream>>>(node_species, W_out_s1, W_o1, W_o2,
                                         res1_sp, n0, out);
}